// FNetBlock_73220602462312
// MI455X (gfx1250) — compile-verified
//
#include <hip/hip_runtime.h>
#include <math.h>

typedef __attribute__((ext_vector_type(2))) float v2f;
typedef __attribute__((ext_vector_type(8))) float v8f;
typedef __attribute__((ext_vector_type(4))) unsigned int u32x4;
typedef __attribute__((ext_vector_type(8))) int i32x8;
typedef __attribute__((ext_vector_type(4))) int i32x4;

#define BATCH 8
#define SEQ   2048
#define DIM   1024

#define BM 128
#define BN 128
#define BK 32
#define APAD 4
#define LDT_S (BK + APAD)  // 36 floats per LDS tile row (A and B^T alike)

#if defined(__has_builtin)
#if __has_builtin(__builtin_amdgcn_tensor_load_to_lds) && \
    __has_builtin(__builtin_amdgcn_s_wait_tensorcnt)
#define USE_TDM 1
#endif
#endif
#ifndef USE_TDM
#define USE_TDM 0
#endif

// ---------------------------------------------------------------------------
// Softmax over the sequence axis (dim=-2). One thread per (b, d) column,
// coalesced across d. Online max+sum pass, then normalize pass.
// ---------------------------------------------------------------------------
__global__ void softmax_seq_kernel(const float* __restrict__ x,
                                   float* __restrict__ vis) {
  const int b = blockIdx.y;
  const int d = blockIdx.x * blockDim.x + threadIdx.x;
  const size_t base = (size_t)b * SEQ * DIM + d;

  float m = -INFINITY;
  float s = 0.0f;
  for (int i = 0; i < SEQ; ++i) {
    float v  = x[base + (size_t)i * DIM];
    float m2 = fmaxf(m, v);
    s = s * __expf(m - m2) + __expf(v - m2);
    m = m2;
  }
  const float inv = 1.0f / s;
  for (int i = 0; i < SEQ; ++i) {
    vis[base + (size_t)i * DIM] = __expf(x[base + (size_t)i * DIM] - m) * inv;
  }
}

// ---------------------------------------------------------------------------
// Twiddle matrices. Exact integer phase reduction: (k*j) mod N with N = 2^p.
// C_D and S_D are symmetric, so they serve directly as their own transpose.
// ---------------------------------------------------------------------------
__global__ void init_dft_dim_kernel(float* __restrict__ CD,
                                    float* __restrict__ SD) {
  const int idx = blockIdx.x * blockDim.x + threadIdx.x;  // DIM*DIM threads
  const int d = idx >> 10;
  const int m = idx & (DIM - 1);
  const int t = (d * m) & (DIM - 1);
  const float ang = (float)t * (6.28318530717958647692f / (float)DIM);
  float sv, cv;
  __sincosf(ang, &sv, &cv);
  CD[idx] = cv;
  SD[idx] = sv;
}

// T2 = [C_S | -S_S], shape SEQ x (2*SEQ), row-major (used as the A operand).
__global__ void init_t2_kernel(float* __restrict__ T2) {
  const size_t idx = (size_t)blockIdx.x * blockDim.x + threadIdx.x;
  const int k  = (int)(idx >> 12);          // / 4096
  const int j  = (int)(idx & (2 * SEQ - 1));
  const int jj = j & (SEQ - 1);
  const int t  = (k * jj) & (SEQ - 1);
  const float ang = (float)t * (6.28318530717958647692f / (float)SEQ);
  float sv, cv;
  __sincosf(ang, &sv, &cv);
  T2[idx] = (j < SEQ) ? cv : -sv;
}

// ---------------------------------------------------------------------------
// Tensor Data Mover: one 2-D tile DMA (global -> LDS) described by a D#.
// Tile: tile_y rows of tile_x fp32 elements, row stride = stride_elems.
// pad codes: interval i -> 2^(i+1) dwords, amount a -> a+1 dwords (LDS only).
// ---------------------------------------------------------------------------
#if USE_TDM
__device__ __forceinline__ void tdm_load_tile(const float* g, const void* lds,
                                              unsigned tile_x, unsigned tile_y,
                                              unsigned long long stride_elems,
                                              unsigned pad_int,
                                              unsigned pad_amt) {
  const unsigned long long ga = (unsigned long long)(uintptr_t)g;
  const unsigned lds_off = (unsigned)(uintptr_t)lds;  // low 32 bits = LDS addr

  u32x4 g0;
  g0[0] = 1u;                                   // count=1, is_restore=0
  g0[1] = lds_off;                              // lds_addr
  g0[2] = (unsigned)ga;                         // global_addr[31:0]
  g0[3] = (unsigned)((ga >> 32) & 0x01FFFFFFull) | (2u << 30);  // type=2

  // tensor dims == tile dims (tiles are always fully in-bounds here)
  i32x8 g1;
  g1[0] = (int)((2u << 16)              // data_size = 4 bytes
                | (1u << 20)            // pad_enable
                | (pad_int << 22) | (pad_amt << 25));
  g1[1] = (int)((tile_x & 0xFFFFu) << 16);                    // tensor_dim0 lo
  g1[2] = (int)((tile_x >> 16) | ((tile_y & 0xFFFFu) << 16)); // d0 hi | d1 lo
  g1[3] = (int)((tile_y >> 16) | (tile_x << 16));             // d1 hi | tile_dim0
  g1[4] = (int)(tile_y & 0xFFFFu);                            // tile_dim1, dim2=0
  g1[5] = (int)(unsigned)(stride_elems & 0xFFFFFFFFull);      // dim0_stride lo
  g1[6] = (int)(unsigned)((stride_elems >> 32) & 0xFFFFull);  // dim0_stride hi
  g1[7] = 0;

  const i32x4 z4 = {};
#if defined(__clang_major__) && (__clang_major__ >= 23)
  const i32x8 z8 = {};
  __builtin_amdgcn_tensor_load_to_lds(g0, g1, z4, z4, z8, 0);
#else
  __builtin_amdgcn_tensor_load_to_lds(g0, g1, z4, z4, 0);
#endif
}
#endif

// ---------------------------------------------------------------------------
// FP32 GEMM via V_WMMA_F32_16X16X4_F32:
//   C = A[MxK] * B[KxN], with B supplied TRANSPOSED (BT is NxK row-major).
// Both A and BT tiles are [row][k] with 32-dword rows -> identical TDM
// descriptors, identical single-b64 LDS fragment loads for A and B operands.
// storeT = 0 : C[row*ldc + col]   (normal, row-major MxN)
// storeT = 1 : C[col*ldc + row]   (write C transposed, NxM row-major)
// Requires M%128==0, N%128==0, K%32==0 (true for all launches below).
// ---------------------------------------------------------------------------
__global__ __launch_bounds__(256)
void gemm_wmma_f32_kernel(const float* __restrict__ A,
                          const float* __restrict__ BT,
                          float* __restrict__ C,
                          int K, int lda, int ldbt, int ldc, int storeT) {
  __shared__ __align__(16) float As[2][BM][LDT_S];  // [m-row][k]
  __shared__ __align__(16) float Bs[2][BN][LDT_S];  // [n-row][k]

  const int tid  = threadIdx.x;
  const int lane = tid & 31;
  const int wid  = tid >> 5;      // 0..7
  const int wm   = wid >> 2;      // 0..1 : wave row    (64 rows each)
  const int wn   = wid & 3;       // 0..3 : wave column (32 cols each)
  const int m0   = blockIdx.y * BM;
  const int n0   = blockIdx.x * BN;
  const int lh   = (lane >> 4) & 1;  // lane >= 16 -> K+2 half / M+8 half
  const int lm   = lane & 15;

  const v8f zero = {};
  v8f acc[4][2];
#pragma unroll
  for (int i = 0; i < 4; ++i)
#pragma unroll
    for (int j = 0; j < 2; ++j) acc[i][j] = zero;

#if USE_TDM
  if (wid == 0) {  // whole wave 0 (EXEC all-ones), uniform descriptors
    tdm_load_tile(A + (size_t)m0 * lda, &As[0][0][0], BK, BM,
                  (unsigned long long)lda, 4u /*32 dw*/, 3u /*4 dw*/);
    tdm_load_tile(BT + (size_t)n0 * ldbt, &Bs[0][0][0], BK, BN,
                  (unsigned long long)ldbt, 4u, 3u);
  }
#endif

  int buf = 0;
  for (int kk = 0; kk < K; kk += BK, buf ^= 1) {
#if USE_TDM
    if (wid == 0) __builtin_amdgcn_s_wait_tensorcnt(0);
    __syncthreads();  // current tile resident; prev compute finished everywhere
    if (wid == 0 && (kk + BK) < K) {
      tdm_load_tile(A + (size_t)m0 * lda + (kk + BK), &As[buf ^ 1][0][0],
                    BK, BM, (unsigned long long)lda, 4u, 3u);
      tdm_load_tile(BT + (size_t)n0 * ldbt + (kk + BK), &Bs[buf ^ 1][0][0],
                    BK, BN, (unsigned long long)ldbt, 4u, 3u);
    }
#else
    // Synchronous fallback: cooperative float4 copies (same layout as TDM).
#pragma unroll
    for (int it = 0; it < (BM * BK) / (256 * 4); ++it) {
      const int idx = tid + 256 * it;  // float4 index, 8 per row
      const int row = idx >> 3;
      const int c4  = (idx & 7) << 2;
      const float4 v = *(const float4*)(A + (size_t)(m0 + row) * lda + kk + c4);
      As[buf][row][c4 + 0] = v.x;
      As[buf][row][c4 + 1] = v.y;
      As[buf][row][c4 + 2] = v.z;
      As[buf][row][c4 + 3] = v.w;
    }
#pragma unroll
    for (int it = 0; it < (BN * BK) / (256 * 4); ++it) {
      const int idx = tid + 256 * it;
      const int row = idx >> 3;
      const int c4  = (idx & 7) << 2;
      const float4 v =
          *(const float4*)(BT + (size_t)(n0 + row) * ldbt + kk + c4);
      Bs[buf][row][c4 + 0] = v.x;
      Bs[buf][row][c4 + 1] = v.y;
      Bs[buf][row][c4 + 2] = v.z;
      Bs[buf][row][c4 + 3] = v.w;
    }
    __syncthreads();
#endif

#pragma unroll
    for (int k2 = 0; k2 < BK; k2 += 4) {
      v2f af[4], bf[2];
      // A frag 16x4: lanes 0-15 hold K={k2,k2+1}, lanes 16-31 K={k2+2,k2+3}
#pragma unroll
      for (int i = 0; i < 4; ++i)
        af[i] = *(const v2f*)&As[buf][wm * 64 + i * 16 + lm][k2 + 2 * lh];
      // B frag 4x16: lane holds column n, same per-lane K split -> single b64
#pragma unroll
      for (int j = 0; j < 2; ++j)
        bf[j] = *(const v2f*)&Bs[buf][wn * 32 + j * 16 + lm][k2 + 2 * lh];
#pragma unroll
      for (int i = 0; i < 4; ++i)
#pragma unroll
        for (int j = 0; j < 2; ++j)
          acc[i][j] = __builtin_amdgcn_wmma_f32_16x16x4_f32(
              false, af[i], false, bf[j], (short)0, acc[i][j], false, false);
    }
#if !USE_TDM
    __syncthreads();
#endif
  }

  // D layout: VGPR r, lane l -> element (r + 8*(l>=16), l&15)
#pragma unroll
  for (int i = 0; i < 4; ++i) {
#pragma unroll
    for (int j = 0; j < 2; ++j) {
      const int rbase = m0 + wm * 64 + i * 16 + lh * 8;
      const int col   = n0 + wn * 32 + j * 16 + lm;
      if (storeT) {
#pragma unroll
        for (int r = 0; r < 8; ++r)
          C[(size_t)col * ldc + (rbase + r)] = acc[i][j][r];
      } else {
#pragma unroll
        for (int r = 0; r < 8; ++r)
          C[(size_t)(rbase + r) * ldc + col] = acc[i][j][r];
      }
    }
  }
}

// ---------------------------------------------------------------------------
extern "C" void kernel_launch(void* const* d_in, const int* in_sizes, int n_in,
                              void* d_out, int out_size, void* d_ws,
                              size_t ws_size, hipStream_t stream) {
  (void)in_sizes; (void)n_in; (void)out_size; (void)ws_size;

  const float* x = (const float*)d_in[0];
  float* out = (float*)d_out;                       // [B,S,D] Re(FFT2)
  float* vis = out + (size_t)BATCH * SEQ * DIM;     // [B,S,D] softmax output

  // Workspace layout (56 MB total):
  float* CD  = (float*)d_ws;                        // DIM x DIM  cos (symmetric)
  float* SD  = CD + (size_t)DIM * DIM;              // DIM x DIM  sin (symmetric)
  float* T2  = SD + (size_t)DIM * DIM;              // SEQ x 2SEQ [C_S | -S_S]
  float* XwT = T2 + (size_t)SEQ * (2 * SEQ);        // DIM x 2SEQ = [A ; B]^T

  softmax_seq_kernel<<<dim3(DIM / 256, BATCH), 256, 0, stream>>>(x, vis);
  init_dft_dim_kernel<<<(DIM * DIM) / 256, 256, 0, stream>>>(CD, SD);
  init_t2_kernel<<<(unsigned)((size_t)SEQ * 2 * SEQ / 256), 256, 0, stream>>>(T2);

  const dim3 gemm_grid(DIM / BN, SEQ / BM);  // (8, 16)
  for (int b = 0; b < BATCH; ++b) {
    const float* Vb = vis + (size_t)b * SEQ * DIM;
    float* Ob = out + (size_t)b * SEQ * DIM;
    // Stage 1: XwT[:, 0:2048]    = (V * C_D)^T   (C_D symmetric => BT = C_D)
    //          XwT[:, 2048:4096] = (V * S_D)^T
    gemm_wmma_f32_kernel<<<gemm_grid, 256, 0, stream>>>(
        Vb, CD, XwT, DIM, DIM, DIM, 2 * SEQ, /*storeT=*/1);
    gemm_wmma_f32_kernel<<<gemm_grid, 256, 0, stream>>>(
        Vb, SD, XwT + SEQ, DIM, DIM, DIM, 2 * SEQ, /*storeT=*/1);
    // Stage 2: out = T2 * [A ; B] ; B operand supplied transposed as XwT
    gemm_wmma_f32_kernel<<<gemm_grid, 256, 0, stream>>>(
        T2, XwT, Ob, 2 * SEQ, 2 * SEQ, 2 * SEQ, DIM, /*storeT=*/0);
  }
}